// MyEmbedding_71347996721877
// MI455X (gfx1250) — compile-verified
//
#include <hip/hip_runtime.h>

#define V_VOCAB  32000
#define DIM      5120
#define T_VIS    32
#define B_BATCH  4
#define S_SEQ    2048
#define VDIM     768

typedef __attribute__((ext_vector_type(2))) float v2f;
typedef __attribute__((ext_vector_type(4))) float v4f;
typedef __attribute__((ext_vector_type(8))) float v8f;

// ---------------------------------------------------------------------------
// Kernel 1: vx[B*T, D] = vision_feats[B*T, VDIM] @ fc_w[VDIM, D] + fc_b
// One wave32 per 16x16 output tile, K-loop of V_WMMA_F32_16X16X4_F32.
//
// f32 WMMA fragment layouts (ISA 7.12.2):
//   A 16x4 : lanes 0-15 -> M=lane,   vgpr0=K0, vgpr1=K1
//            lanes 16-31-> M=lane-16,vgpr0=K2, vgpr1=K3
//   B 4x16 : lanes 0-15 -> N=lane,   vgpr0=K0, vgpr1=K1
//            lanes 16-31-> N=lane-16,vgpr0=K2, vgpr1=K3
//   C/D    : 8 vgprs; lanes 0-15 cover M=0..7, lanes 16-31 cover M=8..15
// ---------------------------------------------------------------------------
__global__ void __launch_bounds__(32) vis_proj_wmma(
    const float* __restrict__ A,    // [128, 768]
    const float* __restrict__ W,    // [768, 5120]
    const float* __restrict__ bias, // [5120]
    float* __restrict__ vx)         // [128, 5120]
{
    const int lane = threadIdx.x;       // 0..31
    const int half = lane >> 4;         // 0 or 1
    const int l16  = lane & 15;
    const int tileN = blockIdx.x;       // 0..319
    const int tileM = blockIdx.y;       // 0..7

    const int m = tileM * 16 + l16;
    const int n = tileN * 16 + l16;

    v8f acc = {};

    // per-lane base pointers; lanes 16-31 own the K+2/K+3 slice
    const float* aRow = A + (size_t)m * VDIM + 2 * half;          // contiguous K pair -> b64 load
    const float* wCol = W + (size_t)(2 * half) * DIM + n;         // two strided b32 loads

#pragma unroll 4
    for (int k0 = 0; k0 < VDIM; k0 += 4) {
        v2f a, b;
        a.x = aRow[k0];
        a.y = aRow[k0 + 1];
        b.x = wCol[(size_t)k0 * DIM];
        b.y = wCol[(size_t)(k0 + 1) * DIM];
        // (neg_a, A, neg_b, B, c_mod, C, reuse_a, reuse_b)
        acc = __builtin_amdgcn_wmma_f32_16x16x4_f32(
            false, a, false, b, (short)0, acc, false, false);
    }

    const float bn = bias[n];
    float* outBase = vx + (size_t)(tileM * 16) * DIM + n;
#pragma unroll
    for (int i = 0; i < 8; ++i) {
        const int mrow = i + 8 * half;                 // C/D layout
        outBase[(size_t)mrow * DIM] = acc[i] + bn;
    }
}

// ---------------------------------------------------------------------------
// Kernel 2: pure-bandwidth gather. One 256-thread block per token row.
// Token is block-uniform -> scalar branch, no divergence. 128-bit loads,
// non-temporal 128-bit stores (output is streamed exactly once; keep L2
// for the hot embedding-table rows and vx).
// ---------------------------------------------------------------------------
__global__ void __launch_bounds__(256) embed_gather(
    const int*   __restrict__ tokens, // [B*S]
    const float* __restrict__ weight, // [V, D]
    const float* __restrict__ fig,    // [2, D]
    const float* __restrict__ vx,     // [B*T, D]
    float*       __restrict__ out)    // [B*S, D]
{
    const int r = blockIdx.x;         // 0..8191
    const int b = r / S_SEQ;
    const int tok = tokens[r];

    const float* src;
    if (tok < V_VOCAB) {
        src = weight + (size_t)tok * DIM;
    } else if (tok < V_VOCAB + 2) {
        src = fig + (size_t)(tok - V_VOCAB) * DIM;
    } else {
        int iv = tok - (V_VOCAB + 2);
        iv = iv < 0 ? 0 : (iv > T_VIS - 1 ? T_VIS - 1 : iv);
        src = vx + ((size_t)b * T_VIS + iv) * DIM;
    }

    const v4f* s4 = (const v4f*)src;
    v4f*       o4 = (v4f*)(out + (size_t)r * DIM);

#pragma unroll
    for (int i = threadIdx.x; i < DIM / 4; i += 256) {   // 5 iterations
        const v4f v = s4[i];
        __builtin_nontemporal_store(v, &o4[i]);
    }
}

// ---------------------------------------------------------------------------
extern "C" void kernel_launch(void* const* d_in, const int* in_sizes, int n_in,
                              void* d_out, int out_size, void* d_ws, size_t ws_size,
                              hipStream_t stream) {
    const int*   tokens = (const int*)  d_in[0];   // [B, S]
    const float* vfeat  = (const float*)d_in[1];   // [B, T, VDIM]
    const float* weight = (const float*)d_in[2];   // [V, D]
    const float* fig    = (const float*)d_in[3];   // [2, D]
    const float* fc_w   = (const float*)d_in[4];   // [VDIM, D]
    const float* fc_b   = (const float*)d_in[5];   // [D]

    float* out = (float*)d_out;                    // [B, S, D]
    float* vx  = (float*)d_ws;                     // B*T*D floats = 2.62 MB scratch

    // 1) vision projection GEMM (WMMA f32)
    dim3 gGemm(DIM / 16, (B_BATCH * T_VIS) / 16);  // 320 x 8 tiles
    vis_proj_wmma<<<gGemm, 32, 0, stream>>>(vfeat, fc_w, fc_b, vx);

    // 2) conditional embedding gather (HBM-bound, ~335 MB traffic)
    embed_gather<<<B_BATCH * S_SEQ, 256, 0, stream>>>(tokens, weight, fig, vx, out);
}